// Encoder_74852690034970
// MI455X (gfx1250) — compile-verified
//
#include <hip/hip_runtime.h>
#include <hip/hip_bf16.h>

#define N_NODES 50000
#define HDIM    96
#define LN_EPS  1e-5f

typedef float v2f __attribute__((ext_vector_type(2)));
typedef float v8f __attribute__((ext_vector_type(8)));

// ---------------------------------------------------------------------------
// Degree / normalization kernels
// ---------------------------------------------------------------------------
__global__ void k_deg_init(float* __restrict__ deg, int n) {
    int i = blockIdx.x * blockDim.x + threadIdx.x;
    if (i < n) deg[i] = 1.0f;   // self-loop contributes 1 to every node's degree
}

__global__ void k_deg_edges(float* __restrict__ deg, const int* __restrict__ dst, int E) {
    int i = blockIdx.x * blockDim.x + threadIdx.x;
    if (i < E) atomicAdd(&deg[dst[i]], 1.0f);
}

__global__ void k_dinv(float* __restrict__ deg, int n) {
    int i = blockIdx.x * blockDim.x + threadIdx.x;
    if (i < n) deg[i] = rsqrtf(deg[i]);   // deg >= 1 always (self loops)
}

// ---------------------------------------------------------------------------
// WMMA fp32 GEMM:  Hout[M x 96] = X[M x K] @ W[K x 96]
// One wave computes one 16x16 tile via V_WMMA_F32_16X16X4_F32.
// B panel (K x 16) for this block's column tile is staged in LDS.
// blockDim = (32, 4): 4 waves -> 4 consecutive M-tiles, blockIdx.y = N-tile.
// ---------------------------------------------------------------------------
__global__ void k_gemm_wmma_f32(const float* __restrict__ X,
                                const float* __restrict__ W,
                                float* __restrict__ Hout,
                                int M, int K) {
    __shared__ float Bs[128 * 16];   // max K = 128

    const int ntile = blockIdx.y;

    // Cooperative load of the K x 16 B-panel into LDS.
    for (int i = threadIdx.y * 32 + threadIdx.x; i < K * 16; i += 128) {
        int k = i >> 4;
        int n = i & 15;
        Bs[i] = W[k * HDIM + ntile * 16 + n];
    }
    __syncthreads();

    const int mtiles = M / 16;
    const int mtile  = blockIdx.x * 4 + threadIdx.y;
    if (mtile >= mtiles) return;                 // wave-uniform: EXEC stays all-1s

    const int lane = threadIdx.x;                // wave32
    const int r    = lane & 15;                  // A: row / B: col / C: col
    const int koff = (lane >> 4) << 1;           // lanes 0-15 -> K{0,1}, 16-31 -> K{2,3}

    const float* __restrict__ arow = X + (size_t)(mtile * 16 + r) * K;

    v8f c = {};
    for (int kk = 0; kk < K; kk += 4) {
        v2f a, b;
        // A fragment: 32-bit A-matrix 16x4 layout (ISA 7.12.2)
        a.x = arow[kk + koff + 0];
        a.y = arow[kk + koff + 1];
        // B fragment: 4x16, rows striped across lanes within a VGPR
        b.x = Bs[(kk + koff + 0) * 16 + r];
        b.y = Bs[(kk + koff + 1) * 16 + r];
        // (neg_a, A, neg_b, B, c_mod, C, reuse_a, reuse_b)
        c = __builtin_amdgcn_wmma_f32_16x16x4_f32(
                false, a, false, b, (short)0, c, false, false);
    }

    // C/D layout: VGPR i -> row (i + (lane>=16)*8), col = lane&15
    const int row0 = mtile * 16 + ((lane >> 4) << 3);
    float* __restrict__ out = Hout + (size_t)row0 * HDIM + ntile * 16 + r;
#pragma unroll
    for (int i = 0; i < 8; ++i) out[(size_t)i * HDIM] = c[i];
}

// ---------------------------------------------------------------------------
// hs[n,f] = h[n,f] * dinv[n]; also seed accumulator (h buffer) with the
// self-loop contribution: agg[n,f] = hs[n,f].
// ---------------------------------------------------------------------------
__global__ void k_prescale(float* __restrict__ h_agg,      // in: h, out: agg seed
                           float* __restrict__ hs,         // out: scaled gather source
                           const float* __restrict__ dinv,
                           int total) {
    int i = blockIdx.x * blockDim.x + threadIdx.x;
    if (i < total) {
        int n = i / HDIM;
        float s = h_agg[i] * dinv[n];
        hs[i]    = s;
        h_agg[i] = s;   // self-loop seed
    }
}

// ---------------------------------------------------------------------------
// Edge scatter: one wave per edge. agg[dst,:] += hs[src,:]
// 96 floats = 3 per lane, coalesced; fp32 global atomics resolve in L2
// (19.2 MB accumulator fits in the 192 MB L2).
// ---------------------------------------------------------------------------
__global__ void k_scatter(const float* __restrict__ hs,
                          float* __restrict__ agg,
                          const int* __restrict__ src,
                          const int* __restrict__ dst,
                          int E) {
    int wid  = (blockIdx.x * blockDim.x + threadIdx.x) >> 5;
    int lane = threadIdx.x & 31;
    if (wid >= E) return;

    int s = src[wid];
    int d = dst[wid];
    const float* __restrict__ hrow = hs  + (size_t)s * HDIM;
    float*       __restrict__ arow = agg + (size_t)d * HDIM;
#pragma unroll
    for (int i = 0; i < 3; ++i) {
        int f = lane + 32 * i;
        atomicAdd(&arow[f], hrow[f]);
    }
}

// ---------------------------------------------------------------------------
// Fused epilogue: y = PReLU(LayerNorm(agg * dinv[n] + bias)) — wave per node,
// wave32 __shfl_xor reductions for mean/variance over the 96 features.
// ---------------------------------------------------------------------------
__global__ void k_epilogue(const float* __restrict__ agg,
                           const float* __restrict__ dinv,
                           const float* __restrict__ bias,
                           const float* __restrict__ gamma,
                           const float* __restrict__ beta,
                           const float* __restrict__ alpha,
                           float* __restrict__ y,
                           int N) {
    int wid  = (blockIdx.x * blockDim.x + threadIdx.x) >> 5;
    int lane = threadIdx.x & 31;
    if (wid >= N) return;

    const float* __restrict__ row = agg + (size_t)wid * HDIM;
    float dv = dinv[wid];

    float v[3];
#pragma unroll
    for (int i = 0; i < 3; ++i) {
        int f = lane + 32 * i;
        v[i] = row[f] * dv + bias[f];
    }

    float s = v[0] + v[1] + v[2];
#pragma unroll
    for (int m = 16; m > 0; m >>= 1) s += __shfl_xor(s, m, 32);
    float mu = s * (1.0f / (float)HDIM);

    float d0 = v[0] - mu, d1 = v[1] - mu, d2 = v[2] - mu;
    float q = d0 * d0 + d1 * d1 + d2 * d2;
#pragma unroll
    for (int m = 16; m > 0; m >>= 1) q += __shfl_xor(q, m, 32);
    float rstd = rsqrtf(q * (1.0f / (float)HDIM) + LN_EPS);

    float dvv[3] = {d0, d1, d2};
#pragma unroll
    for (int i = 0; i < 3; ++i) {
        int f = lane + 32 * i;
        float o = dvv[i] * rstd * gamma[f] + beta[f];
        y[(size_t)wid * HDIM + f] = (o >= 0.0f) ? o : alpha[f] * o;
    }
}

// ---------------------------------------------------------------------------
// Host launcher
// ---------------------------------------------------------------------------
extern "C" void kernel_launch(void* const* d_in, const int* in_sizes, int n_in,
                              void* d_out, int out_size, void* d_ws, size_t ws_size,
                              hipStream_t stream) {
    const float* x   = (const float*)d_in[0];
    const int*   ei  = (const int*)  d_in[1];
    const float* W1  = (const float*)d_in[2];
    const float* b1  = (const float*)d_in[3];
    const float* W2  = (const float*)d_in[4];
    const float* b2  = (const float*)d_in[5];
    const float* g1  = (const float*)d_in[6];
    const float* be1 = (const float*)d_in[7];
    const float* g2  = (const float*)d_in[8];
    const float* be2 = (const float*)d_in[9];
    const float* a   = (const float*)d_in[10];

    const int N   = N_NODES;
    const int Cin = in_sizes[0] / N;        // 128
    const int E   = in_sizes[1] / 2;        // 800000
    const int NH  = N * HDIM;

    const int* src = ei;
    const int* dst = ei + E;

    // Workspace layout (fp32): dinv[N] | bufH[N*H] (gemm out / agg) | bufS[N*H]
    float* dinv = (float*)d_ws;
    float* bufH = dinv + ((N + 255) & ~255);
    float* bufS = bufH + (size_t)NH;
    float* y1   = (float*)d_out;            // layer-1 output lives in d_out, reread by GEMM2

    // --- Degree / normalization (shared by both layers)
    k_deg_init <<<(N + 255) / 256, 256, 0, stream>>>(dinv, N);
    k_deg_edges<<<(E + 255) / 256, 256, 0, stream>>>(dinv, dst, E);
    k_dinv     <<<(N + 255) / 256, 256, 0, stream>>>(dinv, N);

    dim3 gemmBlock(32, 4);
    dim3 gemmGrid1((N / 16 + 3) / 4, HDIM / 16);
    const int scatterGrid = (E * 32 + 255) / 256;
    const int epiGrid     = (N * 32 + 255) / 256;

    // ================= Layer 1 =================
    k_gemm_wmma_f32<<<gemmGrid1, gemmBlock, 0, stream>>>(x, W1, bufH, N, Cin);
    k_prescale<<<(NH + 255) / 256, 256, 0, stream>>>(bufH, bufS, dinv, NH);
    k_scatter <<<scatterGrid, 256, 0, stream>>>(bufS, bufH, src, dst, E);
    k_epilogue<<<epiGrid, 256, 0, stream>>>(bufH, dinv, b1, g1, be1, a, y1, N);

    // ================= Layer 2 =================
    k_gemm_wmma_f32<<<gemmGrid1, gemmBlock, 0, stream>>>(y1, W2, bufH, N, HDIM);
    k_prescale<<<(NH + 255) / 256, 256, 0, stream>>>(bufH, bufS, dinv, NH);
    k_scatter <<<scatterGrid, 256, 0, stream>>>(bufS, bufH, src, dst, E);
    k_epilogue<<<epiGrid, 256, 0, stream>>>(bufH, dinv, b2, g2, be2, a, (float*)d_out, N);
}